// LSTM_24756191494330
// MI455X (gfx1250) — compile-verified
//
#include <hip/hip_runtime.h>
#include <hip/hip_bf16.h>
#include <stdint.h>

typedef __attribute__((ext_vector_type(2))) float v2f;
typedef __attribute__((ext_vector_type(8))) float v8f;

#define IN_SIZE   2048
#define STATE_SZ  4096
#define BATCH     4096

#define APAD 20   // A tile row stride in floats: 16B-aligned (80B) and bank-conflict-free

// fast tanh via v_exp_f32, saturates correctly at +-1 for large |x|
__device__ __forceinline__ float tanh_fast(float v) {
    float e = __expf(-2.0f * v);
    return 2.0f / (1.0f + e) - 1.0f;
}
__device__ __forceinline__ float sigmoid_fast(float v) {
    return 1.0f / (1.0f + __expf(-v));
}

// CDNA5 async copy: global -> LDS, 16 bytes per lane, tracked by ASYNCcnt.
__device__ __forceinline__ void async_copy_b128(uint32_t lds_byte_addr, const float* gaddr) {
    asm volatile("global_load_async_to_lds_b128 %0, %1, off"
                 :: "v"(lds_byte_addr), "v"(gaddr)
                 : "memory");
}
__device__ __forceinline__ void wait_async0() {
    asm volatile("s_wait_asynccnt 0" ::: "memory");
}

// Block tile: 128 (M=state rows) x 64 (N=batch cols), 256 threads = 8 waves.
// Waves arranged 4(M) x 2(N); each wave computes a 32x32 tile for EACH of the
// 4 gates with V_WMMA_F32_16X16X4_F32 (full fp32 precision).
// Double-buffered LDS filled by GLOBAL_LOAD_ASYNC_TO_LDS_B128; one barrier
// per 16-wide K slab; slab s+1 HBM latency overlaps slab s's 64 WMMAs.
__global__ __launch_bounds__(256) void lstm_fused_wmma_f32(
    const float* __restrict__ x,        // (2048, 4096)
    const float* __restrict__ h,        // (4096, 4096) prev_out
    const float* __restrict__ c_prev,   // (4096, 4096) prev_state
    const float* __restrict__ Wf_x, const float* __restrict__ Wf_h, const float* __restrict__ bf,
    const float* __restrict__ Wi_x, const float* __restrict__ Wi_h, const float* __restrict__ bi,
    const float* __restrict__ Wc_x, const float* __restrict__ Wc_h, const float* __restrict__ bc,
    const float* __restrict__ Wo_x, const float* __restrict__ Wo_h, const float* __restrict__ bo,
    float* __restrict__ out_state,      // (4096, 4096)
    float* __restrict__ out_h)          // (4096, 4096)
{
    __shared__ __align__(16) float As[2][4][128][APAD]; // 2 x 40960 B
    __shared__ __align__(16) float Bs[2][16][64];       // 2 x  4096 B

    const int tid  = threadIdx.x;
    const int lane = tid & 31;
    const int lh   = lane & 15;
    const bool hi  = lane >= 16;         // high half-wave holds K+2/K+3 rows
    const int wave = tid >> 5;
    const int wm   = wave & 3;           // 0..3 (M direction)
    const int wn   = wave >> 2;          // 0..1 (N direction)

    const int m0 = blockIdx.y * 128;
    const int n0 = blockIdx.x * 64;

    const float* WX[4] = { Wf_x, Wi_x, Wc_x, Wo_x };
    const float* WH[4] = { Wf_h, Wi_h, Wc_h, Wo_h };

    // per-thread load slots (same for every slab)
    const int arow0 = tid >> 2;          // A rows for rep 0 / rep 1
    const int arow1 = (tid + 256) >> 2;
    const int akq0  = (tid & 3) * 4;     // 0,4,8,12
    const int krow  = tid >> 4;          // B: 0..15
    const int ncol  = (tid & 15) * 4;    // B: 0..60

    v8f acc[4][2][2];
#pragma unroll
    for (int g = 0; g < 4; ++g)
#pragma unroll
        for (int mt = 0; mt < 2; ++mt)
#pragma unroll
            for (int nt = 0; nt < 2; ++nt)
                acc[g][mt][nt] = v8f{};

    const int NS1 = IN_SIZE / 16;                // 128 slabs over x
    const int NSTEPS = NS1 + STATE_SZ / 16;      // + 256 slabs over h = 384

    // issue all 9 async B128 copies for slab s into buffer b
    auto issue_slab = [&](int s, int b) {
        const bool ph = (s >= NS1);
        const int  ks = ph ? (s - NS1) * 16 : s * 16;
        const int  lda = ph ? STATE_SZ : IN_SIZE;
        const float* const* W = ph ? WH : WX;
        const float* Bsrc = ph ? h : x;
#pragma unroll
        for (int g = 0; g < 4; ++g) {
            async_copy_b128((uint32_t)(uintptr_t)&As[b][g][arow0][akq0],
                            W[g] + (size_t)(m0 + arow0) * lda + ks + akq0);
            async_copy_b128((uint32_t)(uintptr_t)&As[b][g][arow1][akq0],
                            W[g] + (size_t)(m0 + arow1) * lda + ks + akq0);
        }
        async_copy_b128((uint32_t)(uintptr_t)&Bs[b][krow][ncol],
                        Bsrc + (size_t)(ks + krow) * BATCH + n0 + ncol);
    };

    issue_slab(0, 0);   // prologue

    for (int s = 0; s < NSTEPS; ++s) {
        const int b = s & 1;
        wait_async0();        // slab s (issued last iteration) has landed in LDS
        __syncthreads();      // all waves: slab s complete, nobody still reads buf b^1
        if (s + 1 < NSTEPS)
            issue_slab(s + 1, b ^ 1);   // overlap with this slab's WMMAs

        const float (*Ab)[128][APAD] = As[b];
        const float (*Bb)[64]        = Bs[b];

#pragma unroll
        for (int kk = 0; kk < 16; kk += 4) {
            const int k0 = kk + (hi ? 2 : 0);
            v2f bfr[2];
#pragma unroll
            for (int nt = 0; nt < 2; ++nt) {
                int n = wn * 32 + nt * 16 + lh;
                bfr[nt][0] = Bb[k0][n];
                bfr[nt][1] = Bb[k0 + 1][n];
            }
#pragma unroll
            for (int g = 0; g < 4; ++g) {
#pragma unroll
                for (int mt = 0; mt < 2; ++mt) {
                    int m = wm * 32 + mt * 16 + lh;
                    v2f af;
                    af[0] = Ab[g][m][k0];
                    af[1] = Ab[g][m][k0 + 1];
#pragma unroll
                    for (int nt = 0; nt < 2; ++nt) {
                        acc[g][mt][nt] = __builtin_amdgcn_wmma_f32_16x16x4_f32(
                            false, af, false, bfr[nt],
                            (short)0, acc[g][mt][nt], false, false);
                    }
                }
            }
        }
    }

    // ---- fused LSTM epilogue ----
    // C/D layout: VGPR r, lanes 0-15 -> (M=r, N=lane); lanes 16-31 -> (M=r+8, N=lane-16)
#pragma unroll
    for (int mt = 0; mt < 2; ++mt) {
#pragma unroll
        for (int nt = 0; nt < 2; ++nt) {
            const int ng = n0 + wn * 32 + nt * 16 + lh;
#pragma unroll
            for (int r = 0; r < 8; ++r) {
                const int mg = m0 + wm * 32 + mt * 16 + r + (hi ? 8 : 0);
                float zf = acc[0][mt][nt][r] + bf[mg];
                float zi = acc[1][mt][nt][r] + bi[mg];
                float zc = acc[2][mt][nt][r] + bc[mg];
                float zo = acc[3][mt][nt][r] + bo[mg];
                float ft = sigmoid_fast(zf);
                float it = sigmoid_fast(zi);
                float ct = tanh_fast(zc);
                float ot = sigmoid_fast(zo);
                size_t idx = (size_t)mg * BATCH + ng;
                float ns = c_prev[idx] * ft + it * ct;
                out_state[idx] = ns;
                out_h[idx]     = tanh_fast(ns) * ot;
            }
        }
    }
}

extern "C" void kernel_launch(void* const* d_in, const int* in_sizes, int n_in,
                              void* d_out, int out_size, void* d_ws, size_t ws_size,
                              hipStream_t stream) {
    const float* x      = (const float*)d_in[0];
    const float* h      = (const float*)d_in[1];
    const float* c_prev = (const float*)d_in[2];
    const float* Wf_x   = (const float*)d_in[3];
    const float* Wf_h   = (const float*)d_in[4];
    const float* bf     = (const float*)d_in[5];
    const float* Wi_x   = (const float*)d_in[6];
    const float* Wi_h   = (const float*)d_in[7];
    const float* bi     = (const float*)d_in[8];
    const float* Wc_x   = (const float*)d_in[9];
    const float* Wc_h   = (const float*)d_in[10];
    const float* bc     = (const float*)d_in[11];
    const float* Wo_x   = (const float*)d_in[12];
    const float* Wo_h   = (const float*)d_in[13];
    const float* bo     = (const float*)d_in[14];

    float* out_state = (float*)d_out;
    float* out_h     = (float*)d_out + (size_t)STATE_SZ * BATCH;

    dim3 grid(BATCH / 64, STATE_SZ / 128);  // (64, 32)
    lstm_fused_wmma_f32<<<grid, 256, 0, stream>>>(
        x, h, c_prev,
        Wf_x, Wf_h, bf,
        Wi_x, Wi_h, bi,
        Wc_x, Wc_h, bc,
        Wo_x, Wo_h, bo,
        out_state, out_h);
}